// SCM_19061064860184
// MI455X (gfx1250) — compile-verified
//
#include <hip/hip_runtime.h>

// Spatial context module: 4 directional 5-tap 1-D conv recurrences (C=1 makes
// each 5x5 conv collapse to its center row/column).  Two launches of ONE scan
// kernel:
//   launch 1: scan over h (taps = kernel row 2), out = ws[b][w][H-1-h]
//   launch 2: scan over w (taps = kernel col 2), reads ws columns coalesced,
//             out[b][h][W-1-t].
// Latency-bound design: 8 WGs (one batch each, one per WGP), 1024 threads =
// 32 wave32s, ONE barrier per recurrence step via fully double-buffered LDS
// state; next input row prefetched with CDNA5 async global->LDS copies
// (ASYNCcnt + s_wait_asynccnt).
// Workspace requirement: 8*1024*1024*4 = 32 MB in d_ws.

#define NDIM  1024
#define BATCH 8

__device__ __forceinline__ float dot5(const float* v, int i, const float c[5]) {
  float r = c[0] * v[i];
  r = fmaf(c[1], v[i + 1], r);
  r = fmaf(c[2], v[i + 2], r);
  r = fmaf(c[3], v[i + 3], r);
  r = fmaf(c[4], v[i + 4], r);
  return r;
}

// CDNA5 async global->LDS copy (ASYNCcnt-tracked), GVS mode:
// each lane copies one dword from the global address (saddr base plus vaddr offset) to its LDS offset in vdst.
__device__ __forceinline__ void async_row_b32(unsigned lds_off,
                                              unsigned long long gbase,
                                              unsigned goff) {
  asm volatile("global_load_async_to_lds_b32 %0, %1, %2"
               :: "v"(lds_off), "v"(goff), "s"(gbase)
               : "memory");
}
__device__ __forceinline__ void wait_async0() {
  asm volatile("s_wait_asynccnt 0" ::: "memory");
}

// Fused double-scan:  s_k = aA*s_{k-1} + bA*x_k   (s_0 = pA*x_0)
//                     u_k = aB*u_{k-1} + bB*s_k   (u_0 = pB*s_0)
// u_k is stored to outb[lane*N + (N-1-k)]  (reverse order).
// tapOff/tapStride: kernel row 2 -> (10,1); kernel col 2 -> (2,5).
__global__ __launch_bounds__(1024) void scm_scan(
    const float* __restrict__ in, float* __restrict__ outp,
    const float* __restrict__ preAw, const float* __restrict__ preAb,
    const float* __restrict__ preBw, const float* __restrict__ preBb,
    const float* __restrict__ convAw, const float* __restrict__ convAb,
    const float* __restrict__ convBw, const float* __restrict__ convBb,
    int tapOff, int tapStride) {
  constexpr int N = NDIM;
  // Ping-pong everything: iteration h reads parity h&1, writes parity (h&1)^1.
  __shared__ float sS[2][N + 4];   // s state (+/-2 zero halo)
  __shared__ float sU[2][N + 4];   // u state (one step behind s)
  __shared__ float sX[2][N + 4];   // input row (row r lives in buffer r&1)

  const int w = threadIdx.x;
  const int b = blockIdx.x;
  const float* inb = in + (size_t)b * N * N;
  float* outb = outp + (size_t)b * N * N + (size_t)w * N;

  // Extract the 10+10 live taps (uniform scalar loads).
  float pA[5], pB[5], aA[5], bA[5], aB[5], bB[5];
#pragma unroll
  for (int j = 0; j < 5; ++j) {
    const int t = tapOff + j * tapStride;
    pA[j] = preAw[t];
    pB[j] = preBw[t];
    aA[j] = convAw[t];       // channel 0: carry
    bA[j] = convAw[25 + t];  // channel 1: new slice
    aB[j] = convBw[t];
    bB[j] = convBw[25 + t];
  }
  const float pAb0 = preAb[0], pBb0 = preBb[0];
  const float cAb0 = convAb[0], cBb0 = convBb[0];

  // Zero-pad halos once ('same' padding).
  if (w < 2) {
#pragma unroll
    for (int p = 0; p < 2; ++p) {
      sS[p][w] = 0.f; sS[p][N + 2 + w] = 0.f;
      sU[p][w] = 0.f; sU[p][N + 2 + w] = 0.f;
      sX[p][w] = 0.f; sX[p][N + 2 + w] = 0.f;
    }
  }

  const unsigned long long gbase = (unsigned long long)(uintptr_t)inb;
  // Wave-relative LDS byte offsets of this lane's input slots.
  const unsigned x0off = (unsigned)(uintptr_t)(const void*)&sX[0][2 + w];
  const unsigned x1off = (unsigned)(uintptr_t)(const void*)&sX[1][2 + w];

  // ---- prologue ----
  // row 0 -> sX[0]
  async_row_b32(x0off, gbase, (unsigned)((0 * N + w) * 4));
  wait_async0();
  __syncthreads();

  // s_0 into write-parity 1 (read by pipeline step h=1); fetch row 1 -> sX[1].
  const float s0 = dot5(sX[0], w, pA) + pAb0;
  sS[1][2 + w] = s0;
  async_row_b32(x1off, gbase, (unsigned)((1 * N + w) * 4));
  wait_async0();
  __syncthreads();

  // pipeline step h=1 (read parity 1): s_1 and u_0; fetch row 2 -> sX[0].
  async_row_b32(x0off, gbase, (unsigned)((2 * N + w) * 4));
  const float u0 = dot5(sS[1], w, pB) + pBb0;
  const float s1 = dot5(sS[1], w, aA) + dot5(sX[1], w, bA) + cAb0;
  outb[N - 1] = u0;
  sS[0][2 + w] = s1;  // write parity 0
  sU[0][2 + w] = u0;
  wait_async0();
  __syncthreads();

  // ---- steady state: iteration h computes s_h and u_{h-1}; 1 barrier/step --
  for (int h = 2; h < N; ++h) {
    const int p = h & 1;  // read parity (also buffer holding row h)
    if (h + 1 < N)        // prefetch row h+1 into the other input buffer
      async_row_b32(p ? x0off : x1off, gbase, (unsigned)(((h + 1) * N + w) * 4));

    const float sh = dot5(sS[p], w, aA) + dot5(sX[p], w, bA) + cAb0;  // s_h
    const float uh = dot5(sU[p], w, aB) + dot5(sS[p], w, bB) + cBb0;  // u_{h-1}
    outb[N - h] = uh;     // index N-1-(h-1)

    sS[p ^ 1][2 + w] = sh;
    sU[p ^ 1][2 + w] = uh;

    wait_async0();        // lane's async copy landed before anyone may read it
    __syncthreads();      // single barrier: publishes parity p^1 for step h+1
  }

  // ---- epilogue: u_{N-1} from parity N&1 == 0 ----
  outb[0] = dot5(sU[0], w, aB) + dot5(sS[0], w, bB) + cBb0;
}

extern "C" void kernel_launch(void* const* d_in, const int* in_sizes, int n_in,
                              void* d_out, int out_size, void* d_ws, size_t ws_size,
                              hipStream_t stream) {
  (void)in_sizes; (void)n_in; (void)out_size; (void)ws_size;
  const float* x     = (const float*)d_in[0];
  const float* pre1w = (const float*)d_in[1];
  const float* pre1b = (const float*)d_in[2];
  const float* pre2w = (const float*)d_in[3];
  const float* pre2b = (const float*)d_in[4];
  const float* pre3w = (const float*)d_in[5];
  const float* pre3b = (const float*)d_in[6];
  const float* pre4w = (const float*)d_in[7];
  const float* pre4b = (const float*)d_in[8];
  const float* c1w   = (const float*)d_in[9];
  const float* c1b   = (const float*)d_in[10];
  const float* c2w   = (const float*)d_in[11];
  const float* c2b   = (const float*)d_in[12];
  const float* c3w   = (const float*)d_in[13];
  const float* c3b   = (const float*)d_in[14];
  const float* c4w   = (const float*)d_in[15];
  const float* c4b   = (const float*)d_in[16];
  float* ws  = (float*)d_ws;   // 32 MB intermediate, transposed layout
  float* out = (float*)d_out;

  // Stage A (passes 1+2): scan over h, taps = kernel row 2 -> off 10, stride 1.
  scm_scan<<<BATCH, NDIM, 0, stream>>>(x, ws, pre1w, pre1b, pre2w, pre2b,
                                       c1w, c1b, c2w, c2b, 10, 1);
  // Stage B (passes 3+4): scan over w, taps = kernel col 2 -> off 2, stride 5.
  scm_scan<<<BATCH, NDIM, 0, stream>>>(ws, out, pre3w, pre3b, pre4w, pre4b,
                                       c3w, c3b, c4w, c4b, 2, 5);
}